// GNN_drug_14370960572451
// MI455X (gfx1250) — compile-verified
//
#include <hip/hip_runtime.h>
#include <math.h>

#define NN   200000
#define EE   400000
#define GG   4096
#define FIN  77
#define DD   128
#define CAT  384
#define ODIM 256
#define BN_EPS 1e-5f

typedef float v2f __attribute__((ext_vector_type(2)));
typedef float v8f __attribute__((ext_vector_type(8)));
typedef unsigned int u32x4 __attribute__((ext_vector_type(4)));
typedef int i32x4 __attribute__((ext_vector_type(4)));
typedef int i32x8 __attribute__((ext_vector_type(8)));

// ---- TDM: load a (tile_h x tile_w) fp32 tile (row stride = stride elems) into LDS ----
__device__ __forceinline__ void tdm_load_tile(const float* gsrc, unsigned lds_byte_off,
                                              unsigned tile_w, unsigned tile_h,
                                              unsigned stride) {
    unsigned long long ga = (unsigned long long)(const void*)gsrc;
    u32x4 g0;
    g0[0] = 1u;                                            // count=1, user mode
    g0[1] = lds_byte_off;                                  // lds_addr
    g0[2] = (unsigned)(ga & 0xffffffffu);                  // global_addr[31:0]
    g0[3] = ((unsigned)(ga >> 32) & 0x01ffffffu) | 0x80000000u; // addr[56:32] | type=2
    i32x8 g1;
    g1[0] = 0x00020000;                                    // data_size = 4B, no flags
    g1[1] = (int)((tile_w & 0xffffu) << 16);               // tensor_dim0[15:0]
    g1[2] = (int)(((tile_w >> 16) & 0xffffu) |             // tensor_dim0[31:16]
                  ((tile_h & 0xffffu) << 16));             // tensor_dim1[15:0]
    g1[3] = (int)(((tile_h >> 16) & 0xffffu) |             // tensor_dim1[31:16]
                  ((tile_w & 0xffffu) << 16));             // tile_dim0
    g1[4] = (int)(tile_h & 0xffffu);                       // tile_dim1 (tile_dim2=0 -> 2D)
    g1[5] = (int)stride;                                   // tensor_dim0_stride[31:0]
    g1[6] = 0;
    g1[7] = 0;
    i32x4 gz = {0, 0, 0, 0};
#if __clang_major__ >= 23
    i32x8 gz8 = {0, 0, 0, 0, 0, 0, 0, 0};
    __builtin_amdgcn_tensor_load_to_lds(g0, g1, gz, gz, gz8, 0);
#else
    __builtin_amdgcn_tensor_load_to_lds(g0, g1, gz, gz, 0);
#endif
}

// ---- pad w1_0 (77x128) -> (128x128) with zero rows ----
__global__ void k_w1pad(const float* __restrict__ w1, float* __restrict__ w1p) {
    int i = blockIdx.x * 256 + threadIdx.x;      // 16384 threads
    int k = i >> 7, c = i & 127;
    w1p[i] = (k < FIN) ? w1[k * DD + c] : 0.0f;
}

__global__ void k_zero(float* __restrict__ p, int n) {
    int i = blockIdx.x * 256 + threadIdx.x;
    if (i < n) p[i] = 0.0f;
}

// ---- pad x (N x 77) into concat block 0 (stride CAT) ----
__global__ void k_padx(const float* __restrict__ x, float* __restrict__ out) {
    int i = blockIdx.x * 256 + threadIdx.x;      // N*128
    int n = i >> 7, c = i & 127;
    out[(long)n * CAT + c] = (c < FIN) ? x[(long)n * FIN + c] : 0.0f;
}

// ---- per-destination edge linked lists (built once; topology fixed across layers) ----
__global__ void k_inithead(int* __restrict__ head) {
    int n = blockIdx.x * 256 + threadIdx.x;
    if (n < NN) head[n] = -1;
}

__global__ void k_build(const int* __restrict__ ei, int* __restrict__ head,
                        int* __restrict__ nxt) {
    int e = blockIdx.x * 256 + threadIdx.x;
    if (e >= EE) return;
    int dst = ei[EE + e];
    nxt[e] = atomicExch(&head[dst], e);
}

// ---- s[n] = h[n] + sum_{e: dst==n} h[src[e]]  (no atomics, coalesced float4) ----
__global__ void k_gather(const float* __restrict__ hp, const int* __restrict__ ei,
                         const int* __restrict__ head, const int* __restrict__ nxt,
                         float* __restrict__ s) {
    int t = blockIdx.x * 256 + threadIdx.x;      // N*32
    int n = t >> 5, q = t & 31;
    int c0 = q * 4;
    float4 acc = *(const float4*)(hp + (long)n * CAT + c0);
    for (int e = head[n]; e >= 0; e = nxt[e]) {
        int src = ei[e];
        const float4 v = *(const float4*)(hp + (long)src * CAT + c0);
        acc.x += v.x; acc.y += v.y; acc.z += v.z; acc.w += v.w;
    }
    *(float4*)(s + (long)n * DD + c0) = acc;
}

// ---- t = relu(s @ w1 + b1), in place over s; A tile staged in LDS via TDM ----
__global__ void k_gemm1(float* __restrict__ s, const float* __restrict__ w,
                        const float* __restrict__ bias) {
    __shared__ float Atile[16 * DD];             // 8KB, LDS offset 0
    int wv = threadIdx.x >> 5, lane = threadIdx.x & 31;
    int row0 = blockIdx.x * 16;
    // dynamically-dead store: makes the TDM-written LDS visible to the optimizer
    if (blockIdx.x == 0x7fffffffu) Atile[threadIdx.x] = (float)row0;
    if (threadIdx.x == 0) {
        tdm_load_tile(s + (long)row0 * DD, 0u, DD, 16u, DD);
        __builtin_amdgcn_s_wait_tensorcnt(0);
    }
    __syncthreads();
    int n   = wv * 16 + (lane & 15);
    int m   = lane & 15;
    int kph = (lane >> 4) * 2;
    float bv = bias[n];
    v8f acc;
#pragma unroll
    for (int r = 0; r < 8; ++r) acc[r] = bv;
    for (int kb = 0; kb < DD; kb += 4) {
        v2f a = *(const v2f*)(Atile + m * DD + kb + kph);
        v2f b;
        b.x = w[(kb + kph) * DD + n];
        b.y = w[(kb + kph + 1) * DD + n];
        acc = __builtin_amdgcn_wmma_f32_16x16x4_f32(false, a, false, b,
                                                    (short)0, acc, false, false);
    }
    int rbase = row0 + ((lane >> 4) ? 8 : 0);
#pragma unroll
    for (int r = 0; r < 8; ++r)
        s[(long)(rbase + r) * DD + n] = fmaxf(acc[r], 0.0f);
}

// ---- h = relu(t @ w2 + b2) -> concat block (stride CAT) + BN partial sums ----
__global__ void k_gemm2(const float* __restrict__ t, const float* __restrict__ w,
                        const float* __restrict__ bias, float* __restrict__ hout,
                        float* __restrict__ sum, float* __restrict__ sumsq) {
    __shared__ float Atile[16 * DD];
    int wv = threadIdx.x >> 5, lane = threadIdx.x & 31;
    int row0 = blockIdx.x * 16;
    if (blockIdx.x == 0x7fffffffu) Atile[threadIdx.x] = (float)row0;
    if (threadIdx.x == 0) {
        tdm_load_tile(t + (long)row0 * DD, 0u, DD, 16u, DD);
        __builtin_amdgcn_s_wait_tensorcnt(0);
    }
    __syncthreads();
    int n   = wv * 16 + (lane & 15);
    int m   = lane & 15;
    int kph = (lane >> 4) * 2;
    float bv = bias[n];
    v8f acc;
#pragma unroll
    for (int r = 0; r < 8; ++r) acc[r] = bv;
    for (int kb = 0; kb < DD; kb += 4) {
        v2f a = *(const v2f*)(Atile + m * DD + kb + kph);
        v2f b;
        b.x = w[(kb + kph) * DD + n];
        b.y = w[(kb + kph + 1) * DD + n];
        acc = __builtin_amdgcn_wmma_f32_16x16x4_f32(false, a, false, b,
                                                    (short)0, acc, false, false);
    }
    int rbase = row0 + ((lane >> 4) ? 8 : 0);
    float ls = 0.0f, lq = 0.0f;
#pragma unroll
    for (int r = 0; r < 8; ++r) {
        float v = fmaxf(acc[r], 0.0f);
        hout[(long)(rbase + r) * CAT + n] = v;
        ls += v; lq += v * v;
    }
    // lanes L and L^16 hold the same output column n
    ls += __shfl_xor(ls, 16, 32);
    lq += __shfl_xor(lq, 16, 32);
    if (lane < 16) {
        atomicAdd(&sum[n], ls);
        atomicAdd(&sumsq[n], lq);
    }
}

// ---- training-mode BN (biased var) applied in place on concat block ----
__global__ void k_bn(float* __restrict__ h, const float* __restrict__ sum,
                     const float* __restrict__ sumsq, const float* __restrict__ gamma,
                     const float* __restrict__ beta) {
    int i = blockIdx.x * 256 + threadIdx.x;      // N*128
    int n = i >> 7, c = i & 127;
    float mu  = sum[c] * (1.0f / NN);
    float var = sumsq[c] * (1.0f / NN) - mu * mu;
    float inv = rsqrtf(var + BN_EPS);
    long  o = (long)n * CAT + c;
    h[o] = (h[o] - mu) * inv * gamma[c] + beta[c];
}

// ---- segment boundaries from sorted batch ----
__global__ void k_bounds(const int* __restrict__ batch, int* __restrict__ start) {
    int n = blockIdx.x * 256 + threadIdx.x;
    if (n >= NN) return;
    int b = batch[n];
    int prev = (n == 0) ? -1 : batch[n - 1];
    if (b != prev)
        for (int g = prev + 1; g <= b; ++g) start[g] = n;
    if (n == NN - 1)
        for (int g = b + 1; g <= GG; ++g) start[g] = NN;
}

// ---- global max pool per graph ----
__global__ void k_pool(const float* __restrict__ cat, const int* __restrict__ start,
                       float* __restrict__ pooled) {
    int g = blockIdx.x;
    int c = threadIdx.x;                          // 384 threads
    int s0 = start[g], s1 = start[g + 1];
    float m = -INFINITY;
    for (int n = s0; n < s1; ++n)
        m = fmaxf(m, cat[(long)n * CAT + c]);
    pooled[(long)g * CAT + c] = m;
}

// ---- out = relu(pooled @ w_emb + b_emb) : [4096,384]x[384,256], A tile via TDM ----
__global__ void k_gemmf(const float* __restrict__ A, const float* __restrict__ w,
                        const float* __restrict__ bias, float* __restrict__ out) {
    __shared__ float Atile[16 * CAT];             // 24KB
    int wv = threadIdx.x >> 5, lane = threadIdx.x & 31;
    int row0 = blockIdx.x * 16;
    if (blockIdx.x == 0x7fffffffu) Atile[threadIdx.x] = (float)row0;
    if (threadIdx.x == 0) {
        tdm_load_tile(A + (long)row0 * CAT, 0u, CAT, 16u, CAT);
        __builtin_amdgcn_s_wait_tensorcnt(0);
    }
    __syncthreads();
    int n   = blockIdx.y * 128 + wv * 16 + (lane & 15);
    int m   = lane & 15;
    int kph = (lane >> 4) * 2;
    float bv = bias[n];
    v8f acc;
#pragma unroll
    for (int r = 0; r < 8; ++r) acc[r] = bv;
    for (int kb = 0; kb < CAT; kb += 4) {
        v2f a = *(const v2f*)(Atile + m * CAT + kb + kph);
        v2f b;
        b.x = w[(kb + kph) * ODIM + n];
        b.y = w[(kb + kph + 1) * ODIM + n];
        acc = __builtin_amdgcn_wmma_f32_16x16x4_f32(false, a, false, b,
                                                    (short)0, acc, false, false);
    }
    int rbase = row0 + ((lane >> 4) ? 8 : 0);
#pragma unroll
    for (int r = 0; r < 8; ++r)
        out[(long)(rbase + r) * ODIM + n] = fmaxf(acc[r], 0.0f);
}

extern "C" void kernel_launch(void* const* d_in, const int* in_sizes, int n_in,
                              void* d_out, int out_size, void* d_ws, size_t ws_size,
                              hipStream_t stream) {
    const float* x     = (const float*)d_in[0];
    const int*   ei    = (const int*)d_in[1];
    const int*   batch = (const int*)d_in[2];
    const float* w_emb = (const float*)d_in[21];
    const float* b_emb = (const float*)d_in[22];

    // workspace carve-up
    float* outcat = (float*)d_ws;                         // N*CAT
    float* sbuf   = outcat + (size_t)NN * CAT;            // N*DD
    float* w1p    = sbuf + (size_t)NN * DD;               // 128*128
    float* stats  = w1p + 16384;                          // 3 * 2 * 128
    float* pooled = stats + 768;                          // G*CAT
    int*   start  = (int*)(pooled + (size_t)GG * CAT);    // G+1
    int*   head   = start + (GG + 1);                     // N
    int*   nxt    = head + NN;                            // E

    k_w1pad<<<64, 256, 0, stream>>>((const float*)d_in[3], w1p);
    k_zero<<<3, 256, 0, stream>>>(stats, 768);
    k_padx<<<(NN * DD) / 256, 256, 0, stream>>>(x, outcat);
    k_inithead<<<(NN + 255) / 256, 256, 0, stream>>>(head);
    k_build<<<(EE + 255) / 256, 256, 0, stream>>>(ei, head, nxt);

    for (int l = 0; l < 3; ++l) {
        int base = 3 + 6 * l;
        const float* w1 = (l == 0) ? w1p : (const float*)d_in[base + 0];
        const float* b1 = (const float*)d_in[base + 1];
        const float* w2 = (const float*)d_in[base + 2];
        const float* b2 = (const float*)d_in[base + 3];
        const float* ga = (const float*)d_in[base + 4];
        const float* be = (const float*)d_in[base + 5];

        const float* hprev = outcat + ((l == 0) ? 0 : (size_t)(l - 1) * DD);
        float* hout  = outcat + (size_t)l * DD;
        float* sum   = stats + (size_t)l * 256;
        float* sumsq = sum + 128;

        k_gather<<<(NN * 32) / 256, 256, 0, stream>>>(hprev, ei, head, nxt, sbuf);
        k_gemm1<<<NN / 16, 256, 0, stream>>>(sbuf, w1, b1);
        k_gemm2<<<NN / 16, 256, 0, stream>>>(sbuf, w2, b2, hout, sum, sumsq);
        k_bn<<<(NN * DD) / 256, 256, 0, stream>>>(hout, sum, sumsq, ga, be);
    }

    k_bounds<<<(NN + 255) / 256, 256, 0, stream>>>(batch, start);
    k_pool<<<GG, CAT, 0, stream>>>(outcat, start, pooled);
    k_gemmf<<<dim3(GG / 16, 2), 256, 0, stream>>>(pooled, w_emb, b_emb, (float*)d_out);
}